// BoundaryLoss_20710332302051
// MI455X (gfx1250) — compile-verified
//
#include <hip/hip_runtime.h>

// Boundary loss for MI455X (gfx1250, wave32).
//   pred:   [2,5,64,64,64] fp32 logits   (d_in[0])
//   target: [2,64,64,64]   int32 labels  (d_in[1])
//   out:    scalar fp32 mean(|softmax(pred)[:,1:] - onehot| * exp(-sdist^2/50))
//
// Core cost is the separable exact EDT: d[i] = min_j f[j] + (i-j)^2 — a
// (min,+) tropical matmul, not expressible on the matrix cores, so it runs as
// LDS-staged VALU fma+min. WMMA (V_WMMA_F32_16X16X4_F32, ones x partials) is
// used for the exact f32 wave-level reduction of the final loss sum.

#define NN 64
#define N3 (NN * NN * NN)      // 262144 = 2^18
#define NVOL 8                 // B * (C-1)
#define NFIELD 16              // bg + fg distance fields (8 volumes each)
#define BIGF 1e10f

typedef float v2f __attribute__((ext_vector_type(2)));
typedef float v8f __attribute__((ext_vector_type(8)));

// ---------------------------------------------------------------- init masks
// f_bg = 0 where voxel is background of class c (BIG elsewhere)  -> EDT to bg
// f_fg = 0 where voxel is foreground of class c (BIG elsewhere)  -> EDT to fg
__global__ void bl_init(const int* __restrict__ target,
                        float* __restrict__ fbg, float* __restrict__ ffg,
                        float* __restrict__ out) {
  int idx = blockIdx.x * blockDim.x + threadIdx.x;
  if (idx == 0) out[0] = 0.0f;               // harness poisons d_out; re-zero
  if (idx >= NVOL * N3) return;
  int vol = idx >> 18;                       // / N3
  int v   = idx & (N3 - 1);
  int b   = vol >> 2;
  int cls = (vol & 3) + 1;
  bool fg = (target[b * N3 + v] == cls);
  fbg[idx] = fg ? BIGF : 0.0f;
  ffg[idx] = fg ? 0.0f : BIGF;
}

// ---------------------------------------------------- DT pass along W (fast)
// 4 lines of 64 per 256-thread block; line staged in LDS, in-place rewrite.
__global__ void bl_dt_w(float* __restrict__ f) {
  __shared__ float s[4][NN];
  int tid = threadIdx.x;
  int sub = tid >> 6;
  int i   = tid & 63;
  float* p = f + ((size_t)blockIdx.x * 4 + sub) * NN;
  s[sub][i] = p[i];
  __syncthreads();
  float best = 3.0e38f;
#pragma unroll 8
  for (int j = 0; j < NN; ++j) {
    float dij = (float)(i - j);
    best = fminf(best, fmaf(dij, dij, s[sub][j]));
  }
  p[i] = best;
}

// -------------------------------------------------- DT pass along H (stride 64)
// One (vol,d) slab of 64x64 per block, staged in 16KB LDS.
__global__ void bl_dt_h(float* __restrict__ f) {
  __shared__ float tile[NN * NN];            // [h][w]
  int tid = threadIdx.x;
  int vol = blockIdx.x >> 6;
  int d   = blockIdx.x & 63;
  float* base = f + (size_t)vol * N3 + (size_t)d * (NN * NN);
  for (int k = tid; k < NN * NN; k += 256) tile[k] = base[k];
  __syncthreads();
  int w = tid & 63;
  for (int i = tid >> 6; i < NN; i += 4) {
    float best = 3.0e38f;
#pragma unroll 8
    for (int j = 0; j < NN; ++j) {
      float dij = (float)(i - j);
      best = fminf(best, fmaf(dij, dij, tile[j * NN + w]));
    }
    base[i * NN + w] = best;                 // global write; reads are LDS-only
  }
}

// -------------------------------------------------- DT pass along D (stride 4096)
__global__ void bl_dt_d(float* __restrict__ f) {
  __shared__ float tile[NN * NN];            // [d][w]
  int tid = threadIdx.x;
  int vol = blockIdx.x >> 6;
  int h   = blockIdx.x & 63;
  float* base = f + (size_t)vol * N3 + (size_t)h * NN;
  for (int k = tid; k < NN * NN; k += 256)
    tile[k] = base[(size_t)(k >> 6) * (NN * NN) + (k & 63)];
  __syncthreads();
  int w = tid & 63;
  for (int i = tid >> 6; i < NN; i += 4) {
    float best = 3.0e38f;
#pragma unroll 8
    for (int j = 0; j < NN; ++j) {
      float dij = (float)(i - j);
      best = fminf(best, fmaf(dij, dij, tile[j * NN + w]));
    }
    base[(size_t)i * (NN * NN) + w] = best;
  }
}

// ------------------------------------------- fused softmax/weight/loss + reduce
// Grid-stride with exact division (512 blk x 256 thr x 4 iters == 2*N3) so
// EXEC is all-ones at the WMMA (ISA 7.12 requirement).
__global__ void bl_loss(const float* __restrict__ pred,
                        const int* __restrict__ target,
                        const float* __restrict__ fbg,
                        const float* __restrict__ ffg,
                        float* __restrict__ out) {
  const int total = 2 * N3;
  float acc = 0.0f;
  for (int idx = blockIdx.x * blockDim.x + threadIdx.x; idx < total;
       idx += gridDim.x * blockDim.x) {
    int b = idx >> 18;
    int v = idx & (N3 - 1);
    const float* pb = pred + (size_t)b * 5 * N3 + v;
    float p0 = pb[0 * N3], p1 = pb[1 * N3], p2 = pb[2 * N3],
          p3 = pb[3 * N3], p4 = pb[4 * N3];
    float m = fmaxf(fmaxf(fmaxf(p0, p1), fmaxf(p2, p3)), p4);
    float e0 = __expf(p0 - m), e1 = __expf(p1 - m), e2 = __expf(p2 - m),
          e3 = __expf(p3 - m), e4 = __expf(p4 - m);
    float inv = 1.0f / (e0 + e1 + e2 + e3 + e4);
    int lbl = target[idx];
    float ec[4] = {e1, e2, e3, e4};
#pragma unroll
    for (int c = 0; c < 4; ++c) {
      size_t fi = (size_t)(b * 4 + c) * N3 + v;
      float dist = __fsqrt_rn(fbg[fi]) - __fsqrt_rn(ffg[fi]);  // signed EDT
      float wgt  = __expf(dist * dist * (-1.0f / (2.0f * 5.0f * 5.0f)));
      float tgt  = (lbl == c + 1) ? 1.0f : 0.0f;
      acc += fabsf(fmaf(ec[c], inv, -tgt)) * wgt;
    }
  }

  // ---- wave32 reduction via V_WMMA_F32_16X16X4_F32 (ones * partials) ----
  // B(4x16) holds lane partials in one VGPR, other VGPR = 0; A = ones.
  // D[i,j] = sum_k B[k,j] = p[j] + p[j+16] (any two-row-per-VGPR pairing).
  v2f a;  a.x = 1.0f; a.y = 1.0f;
  v2f bm; bm.x = acc; bm.y = 0.0f;
  v8f cm = {};
  cm = __builtin_amdgcn_wmma_f32_16x16x4_f32(false, a, false, bm,
                                             (short)0, cm, false, false);
  float s = cm[0];                           // every lane: p[L&15]+p[(L&15)+16]
  s += __shfl_xor(s, 1, 32);
  s += __shfl_xor(s, 2, 32);
  s += __shfl_xor(s, 4, 32);
  s += __shfl_xor(s, 8, 32);                 // full wave sum in every lane

  __shared__ float wsum[8];
  int lane = threadIdx.x & 31;
  int wid  = threadIdx.x >> 5;
  if (lane == 0) wsum[wid] = s;
  __syncthreads();
  if (threadIdx.x == 0) {
    float t = 0.0f;
#pragma unroll
    for (int i = 0; i < 8; ++i) t += wsum[i];
    atomicAdd(out, t * (1.0f / (float)(NVOL * N3)));   // global mean
  }
}

// ------------------------------------------------------------------- launcher
extern "C" void kernel_launch(void* const* d_in, const int* in_sizes, int n_in,
                              void* d_out, int out_size, void* d_ws, size_t ws_size,
                              hipStream_t stream) {
  const float* pred   = (const float*)d_in[0];
  const int*   target = (const int*)d_in[1];
  float* out = (float*)d_out;
  float* fbg = (float*)d_ws;                 // 8 MB
  float* ffg = fbg + (size_t)NVOL * N3;      // 8 MB (contiguous: 16 fields)

  bl_init<<<(NVOL * N3) / 256, 256, 0, stream>>>(target, fbg, ffg, out);
  // 16 fields * 64*64 lines / 4 lines-per-block
  bl_dt_w<<<(NFIELD * NN * NN) / 4, 256, 0, stream>>>(fbg);
  bl_dt_h<<<NFIELD * NN, 256, 0, stream>>>(fbg);
  bl_dt_d<<<NFIELD * NN, 256, 0, stream>>>(fbg);
  bl_loss<<<512, 256, 0, stream>>>(pred, target, fbg, ffg, out);
}